// Distance_Module_68410239091192
// MI455X (gfx1250) — compile-verified
//
#include <hip/hip_runtime.h>
#include <float.h>

#if defined(__HIP_DEVICE_COMPILE__) && !__has_builtin(__builtin_amdgcn_wmma_f32_16x16x4_f32)
#error "CDNA5: __builtin_amdgcn_wmma_f32_16x16x4_f32 not available on this toolchain"
#endif

typedef __attribute__((ext_vector_type(2))) float f32x2;
typedef __attribute__((ext_vector_type(8))) float f32x8;
typedef __attribute__((ext_vector_type(4))) unsigned int u32x4;
typedef __attribute__((ext_vector_type(8))) unsigned int u32x8;

#define NB   512   // batch
#define NT   77    // text rows
#define NF   12    // video rows
#define ND   512   // feature dim
#define MT   5     // ceil(77/16) text tiles

// ---------------------------------------------------------------------------
// TDM: issue tensor_load_to_lds for a 2D tile (tile = tile_rows x 512 f32,
// tensor = rows_tensor x 512 f32, row stride 512). Rows beyond rows_tensor are
// zero-filled by TDM OOB handling (ISA 08 §8.2). Descriptor per ISA §8.3/8.4.
// Uniform args only -> descriptor lives in SGPRs as the instruction requires.
// ---------------------------------------------------------------------------
__device__ __attribute__((always_inline))
void tdm_load_2d(unsigned lds_off, const float* gptr,
                 unsigned rows_tensor, unsigned tile_rows) {
    unsigned long long ga = (unsigned long long)(uintptr_t)gptr;
    u32x4 g0;
    g0[0] = 1u;                                   // count=1 (valid), user mode
    g0[1] = lds_off;                              // lds_addr (byte offset)
    g0[2] = (unsigned)(ga & 0xFFFFFFFFu);         // global_addr[31:0]
    g0[3] = (unsigned)((ga >> 32) & 0x01FFFFFFu)  // global_addr[56:32]
          | (2u << 30);                           // type=2 ("image")
    u32x8 g1;
    g1[0] = (2u << 16);            // workgroup_mask=0, data_size=2 (4 bytes)
    g1[1] = (ND & 0xFFFFu) << 16;  // tensor_dim0[15:0] @ bits 63:48
    g1[2] = (rows_tensor & 0xFFFFu) << 16;  // tensor_dim0 hi=0; tensor_dim1 lo
    g1[3] = (ND & 0xFFFFu) << 16;  // tensor_dim1 hi=0; tile_dim0=512
    g1[4] = tile_rows & 0xFFFFu;   // tile_dim1; tile_dim2=0 (unused)
    g1[5] = ND;                    // tensor_dim0_stride[31:0] = 512 elements
    g1[6] = 0u;                    // stride hi16 = 0; tensor_dim1_stride lo = 0
    g1[7] = 0u;
    asm volatile("tensor_load_to_lds %0, %1" :: "s"(g0), "s"(g1) : "memory");
}

// ---------------------------------------------------------------------------
// Kernel 1: per-batch dmin/dmax of d = 1 - cos(t_m, v_n) via f32 WMMA.
// One workgroup (256 thr = 8 waves) per batch. TDM double-buffers text tiles
// into LDS while the previous tile is consumed; video staged once.
// Raw G = T*V^T accumulated per wave over a 64-wide K slice, merged with
// ds_add_f32, then scaled by inverse norms while scanning for min/max.
// ---------------------------------------------------------------------------
__global__ __launch_bounds__(256)
void k_pair_minmax(const float* __restrict__ text,
                   const float* __restrict__ video,
                   float* __restrict__ dmin_g,
                   float* __restrict__ dmax_g) {
    __shared__ float tlds[2][16 * ND];  // 2 x 32 KB text tiles (row-major)
    __shared__ float vlds[16 * ND];     // 32 KB video (rows 12-15 TDM-zeroed)
    __shared__ float glds[16 * 16];     // merged raw dot products
    __shared__ float nrm[16];           // text row sq-norms (per tile)
    __shared__ float vnrm[16];          // video row sq-norms
    __shared__ float invT[16];
    __shared__ float invV[16];
    __shared__ float redmin[8], redmax[8];
    __shared__ float sdmin, sdmax;

    const int b    = blockIdx.x;
    const int tid  = threadIdx.x;
    const int lane = tid & 31;
    const int wave = tid >> 5;
    const int lm   = lane & 15;   // M for A-frag, N for B-frag / C-frag
    const int hi   = lane >> 4;   // 0: lanes 0-15, 1: lanes 16-31

    const float* tg = text  + (size_t)b * NT * ND;
    const float* vg = video + (size_t)b * NF * ND;

    const unsigned t_off0 = (unsigned)(uintptr_t)&tlds[0][0];
    const unsigned t_off1 = (unsigned)(uintptr_t)&tlds[1][0];
    const unsigned v_off  = (unsigned)(uintptr_t)&vlds[0];

    // ---- prologue: kick off video + first text tile DMA (wave 0 only) ----
    if (wave == 0) {
        tdm_load_2d(v_off,  vg, NF, 16);   // 12 valid rows, 4 zero-filled
        tdm_load_2d(t_off0, tg, NT, 16);
    }
    if (tid == 0) { sdmin = FLT_MAX; sdmax = -FLT_MAX; }

    for (int mt = 0; mt < MT; ++mt) {
        const int rows = min(16, NT - mt * 16);
        if (wave == 0) {
            if (mt + 1 < MT) {
                // overlap: fetch next tile into the other buffer
                tdm_load_2d((mt + 1) & 1 ? t_off1 : t_off0,
                            tg + (size_t)(mt + 1) * 16 * ND,
                            (unsigned)(NT - (mt + 1) * 16), 16);
                asm volatile("s_wait_tensorcnt 0x1" ::: "memory");
            } else {
                asm volatile("s_wait_tensorcnt 0x0" ::: "memory");
            }
        }
        glds[tid] = 0.0f;
        if (tid < 16) nrm[tid] = 0.0f;
        __syncthreads();   // DMA for tile mt (and video) complete for everyone

        const float* tb = &tlds[mt & 1][0];

        // ---- row sq-norms from LDS: thread (r, seg) sums 32 elements ----
        {
            const int r = tid >> 4, seg = tid & 15;
            const float* p = tb + r * ND + seg * 32;
            float s = 0.0f;
            #pragma unroll
            for (int i = 0; i < 32; ++i) s += p[i] * p[i];
            atomicAdd(&nrm[r], s);                    // ds_add_f32
            if (mt == 0) {
                const float* q = vlds + r * ND + seg * 32;
                float sv = 0.0f;
                #pragma unroll
                for (int i = 0; i < 32; ++i) sv += q[i] * q[i];
                atomicAdd(&vnrm[r], sv);
            }
        }
        __syncthreads();
        if (tid < 16) invT[tid] = (tid < rows) ? rsqrtf(nrm[tid]) : 0.0f;
        if (mt == 0 && tid < 16) invV[tid] = (tid < NF) ? rsqrtf(vnrm[tid]) : 0.0f;

        // ---- WMMA chain: wave w owns K in [w*64, w*64+64), 16 steps of K=4 ----
        f32x8 acc = {};
        const int kslice = wave * 64;
        #pragma unroll
        for (int ks = 0; ks < 16; ++ks) {
            const int k = kslice + ks * 4 + hi * 2;  // A-frag: lanes>=16 get K+2,K+3
            const f32x2 a  = *(const f32x2*)(tb   + lm * ND + k);  // ds_load_b64
            const f32x2 bf = *(const f32x2*)(vlds + lm * ND + k);  // ds_load_b64
            acc = __builtin_amdgcn_wmma_f32_16x16x4_f32(
                    /*neg_a=*/false, a, /*neg_b=*/false, bf,
                    /*c_mod=*/(short)0, acc, /*reuse_a=*/false, /*reuse_b=*/false);
        }
        // C/D layout: VGPR i -> M = i + 8*hi, N = lane%16. Merge wave K-slices.
        #pragma unroll
        for (int i = 0; i < 8; ++i)
            atomicAdd(&glds[(i + 8 * hi) * 16 + lm], acc[i]);
        __syncthreads();

        // ---- scale by inv-norms, track min/max of d = 1 - cos ----
        const int tm = tid >> 4, tn = tid & 15;
        const bool valid = (tm < rows) && (tn < NF);
        const float dval = 1.0f - glds[tid] * invT[tm] * invV[tn];
        float vmn = valid ? dval :  FLT_MAX;
        float vmx = valid ? dval : -FLT_MAX;
        #pragma unroll
        for (int off = 16; off > 0; off >>= 1) {
            vmn = fminf(vmn, __shfl_down(vmn, off, 32));
            vmx = fmaxf(vmx, __shfl_down(vmx, off, 32));
        }
        if (lane == 0) { redmin[wave] = vmn; redmax[wave] = vmx; }
        __syncthreads();
        if (tid == 0) {
            float a = sdmin, c = sdmax;
            #pragma unroll
            for (int w = 0; w < 8; ++w) {
                a = fminf(a, redmin[w]);
                c = fmaxf(c, redmax[w]);
            }
            sdmin = a; sdmax = c;
        }
        __syncthreads();  // also guards glds/nrm re-zero + buffer reuse next iter
    }
    if (tid == 0) { dmin_g[b] = sdmin; dmax_g[b] = sdmax; }
}

// ---------------------------------------------------------------------------
// Kernel 2: global mn = min(dmin), mx = max(dmax); store mn and 1/(mx-mn).
// ---------------------------------------------------------------------------
__global__ __launch_bounds__(256)
void k_minmax(const float* __restrict__ dmin_g,
              const float* __restrict__ dmax_g,
              float* __restrict__ mnsc) {
    __shared__ float smin[256], smax[256];
    const int t = threadIdx.x;
    smin[t] = fminf(dmin_g[t], dmin_g[t + 256]);
    smax[t] = fmaxf(dmax_g[t], dmax_g[t + 256]);
    __syncthreads();
    for (int s = 128; s > 0; s >>= 1) {
        if (t < s) {
            smin[t] = fminf(smin[t], smin[t + s]);
            smax[t] = fmaxf(smax[t], smax[t + s]);
        }
        __syncthreads();
    }
    if (t == 0) {
        const float mn = smin[0], mx = smax[0];
        mnsc[0] = mn;
        mnsc[1] = 1.0f / (mx - mn);
    }
}

// ---------------------------------------------------------------------------
// Kernel 3: out[i][j] = ((i==j ? dmin[i] : dmax[i]) - mn) * scale
// ---------------------------------------------------------------------------
__global__ __launch_bounds__(256)
void k_fill(const float* __restrict__ dmin_g,
            const float* __restrict__ dmax_g,
            const float* __restrict__ mnsc,
            float* __restrict__ out) {
    const int i = blockIdx.x;
    const float mn = mnsc[0], sc = mnsc[1];
    const float dmn = dmin_g[i], dmx = dmax_g[i];
    for (int j = threadIdx.x; j < NB; j += 256)
        out[(size_t)i * NB + j] = (((j == i) ? dmn : dmx) - mn) * sc;
}

// ---------------------------------------------------------------------------
extern "C" void kernel_launch(void* const* d_in, const int* in_sizes, int n_in,
                              void* d_out, int out_size, void* d_ws, size_t ws_size,
                              hipStream_t stream) {
    (void)in_sizes; (void)n_in; (void)out_size; (void)ws_size;
    const float* text  = (const float*)d_in[0];   // [512,77,512] f32
    const float* video = (const float*)d_in[1];   // [512,12,512] f32
    float* out  = (float*)d_out;                  // [512,512] f32
    float* ws   = (float*)d_ws;
    float* dmin = ws;          // 512 floats
    float* dmax = ws + NB;     // 512 floats
    float* mnsc = ws + 2 * NB; // 2 floats

    k_pair_minmax<<<NB, 256, 0, stream>>>(text, video, dmin, dmax);
    k_minmax<<<1, 256, 0, stream>>>(dmin, dmax, mnsc);
    k_fill<<<NB, 256, 0, stream>>>(dmin, dmax, mnsc, out);
}